// HierarchicalLLMRec_85676007620773
// MI455X (gfx1250) — compile-verified
//
#include <hip/hip_runtime.h>
#include <hip/hip_bf16.h>

// -------------------------------------------------------------------------
// HierarchicalLLMRec: gather + masked mean-pool, then [64x512]@[512x100000]+b
// MI455X reasoning: GEMM reads proj_w (204.8 MB) exactly once -> HBM-bound
// (~11us floor @ 23.3 TB/s). Compute (6.55 GFLOP) is negligible even with
// f32 WMMA 16x16x4, so we keep full fp32 precision and spend effort on the
// data-movement path: NT loads for the streamed weight matrix, NT stores for
// the streamed logits, temporal loads for the tiny highly-reused user_repr.
// -------------------------------------------------------------------------

#define N_ITEMS 100000
#define HIDDEN  512
#define BATCH   64
#define SEQ     200
#define N_TILES (N_ITEMS / 16)   // 6250, exact

typedef __attribute__((ext_vector_type(2))) float v2f;
typedef __attribute__((ext_vector_type(8))) float v8f;

// ---------------- Phase 1: masked mean pooling -> user_repr [64,512] -------
__global__ __launch_bounds__(256)
void pool_kernel(const int* __restrict__ ids,
                 const float* __restrict__ emb,
                 float* __restrict__ ur) {
    const int b = blockIdx.x;          // batch row
    const int t = threadIdx.x;         // hidden dims t and t+256
    const int* row = ids + b * SEQ;

    float acc0 = 0.0f, acc1 = 0.0f;
    int cnt = 0;
    for (int s = 0; s < SEQ; ++s) {
        const int id = row[s];                       // broadcast load
        const bool valid = (unsigned)id < (unsigned)N_ITEMS;
        cnt += valid ? 1 : 0;
        if (valid) {
            const float* e = emb + (size_t)id * HIDDEN;
            acc0 += e[t];
            acc1 += e[t + 256];
        }
    }
    const float denom = (float)(cnt > 0 ? cnt : 1);
    ur[b * HIDDEN + t]       = acc0 / denom;
    ur[b * HIDDEN + t + 256] = acc1 / denom;
}

// ---------------- Phase 2: WMMA f32 GEMM  logits = ur @ W + bias -----------
// One wave -> one 16-wide N tile, all 4 M tiles (M=64). K loop: 512/4 = 128
// V_WMMA_F32_16X16X4_F32 issues per accumulator set.
//
// ISA fragment layouts (CDNA5 ISA):
//   A 16x4 : lane l (M = l%16), VGPR v : K = v + (l>=16 ? 2 : 0)
//   B 4x16 : VGPR v, lanes 0-15 -> row K=v, lanes 16-31 -> row K=v+2, N=l%16
//   C/D    : VGPR v, lanes 0-15 -> M=v,   lanes 16-31 -> M=v+8,       N=l%16
__global__ __launch_bounds__(256)
void proj_kernel(const float* __restrict__ ur,
                 const float* __restrict__ W,      // [HIDDEN, N_ITEMS] row-major
                 const float* __restrict__ bias,   // [N_ITEMS]
                 float* __restrict__ out) {        // [BATCH, N_ITEMS]
    const int lane  = threadIdx.x & 31;
    const int wave  = threadIdx.x >> 5;
    const int tile  = blockIdx.x * 8 + wave;       // 8 waves per block
    if (tile >= N_TILES) return;                   // wave-uniform: EXEC stays full

    const int n0    = tile * 16;
    const int ncol  = lane & 15;                   // N within tile
    const int khalf = (lane >> 4) << 1;            // 0 for lanes 0-15, 2 for 16-31
    const int mrow  = lane & 15;                   // A-fragment row

    // Fused bias: C/D element (v, lane) sits at column n0+ncol for every v.
    const float bval = bias[n0 + ncol];
    v8f acc[4];
#pragma unroll
    for (int mt = 0; mt < 4; ++mt)
#pragma unroll
        for (int v = 0; v < 8; ++v)
            acc[mt][v] = bval;

    const float* wp_base = W + (size_t)khalf * N_ITEMS + (n0 + ncol);

#pragma unroll 2
    for (int kb = 0; kb < HIDDEN; kb += 4) {
        // B fragment: streamed weights, read once -> non-temporal.
        const float* wp = wp_base + (size_t)kb * N_ITEMS;
        v2f bfrag;
        bfrag.x = __builtin_nontemporal_load(wp);            // row kb+khalf
        bfrag.y = __builtin_nontemporal_load(wp + N_ITEMS);  // row kb+khalf+1

#pragma unroll
        for (int mt = 0; mt < 4; ++mt) {
            // A fragment: tiny (128 KB total), massively reused -> cached loads.
            const float* ap = ur + (size_t)(mt * 16 + mrow) * HIDDEN + kb + khalf;
            v2f afrag;
            afrag.x = ap[0];
            afrag.y = ap[1];
            acc[mt] = __builtin_amdgcn_wmma_f32_16x16x4_f32(
                false, afrag, false, bfrag, (short)0, acc[mt], false, false);
        }
    }

    // Store D: row = mt*16 + v + (lane>=16 ? 8 : 0), col = n0 + ncol.
    const int mhi = (lane >> 4) << 3;
#pragma unroll
    for (int mt = 0; mt < 4; ++mt) {
#pragma unroll
        for (int v = 0; v < 8; ++v) {
            const int m = mt * 16 + v + mhi;
            __builtin_nontemporal_store(acc[mt][v],
                                        out + (size_t)m * N_ITEMS + n0 + ncol);
        }
    }
}

// ---------------------------------------------------------------------------
extern "C" void kernel_launch(void* const* d_in, const int* in_sizes, int n_in,
                              void* d_out, int out_size, void* d_ws, size_t ws_size,
                              hipStream_t stream) {
    const int*   ids  = (const int*)  d_in[0];   // [64,200] int32
    const float* emb  = (const float*)d_in[1];   // [100000,512] f32
    const float* W    = (const float*)d_in[2];   // [512,100000] f32
    const float* bias = (const float*)d_in[3];   // [100000] f32
    float*       out  = (float*)d_out;           // [64,100000] f32
    float*       ur   = (float*)d_ws;            // [64,512] f32 scratch

    pool_kernel<<<BATCH, 256, 0, stream>>>(ids, emb, ur);

    const int blocks = (N_TILES + 7) / 8;        // 8 waves (tiles) per block
    proj_kernel<<<blocks, 256, 0, stream>>>(ur, W, bias, out);
}